// EquivariantAttention_18167711662489
// MI455X (gfx1250) — compile-verified
//
#include <hip/hip_runtime.h>
#include <hip/hip_bf16.h>

// ---------------------------------------------------------------------------
// CDNA5 (gfx1250) fused EquivariantAttention.
//   K1 (edge): WMMA fp32 16x16x4 MLP (32->64->768) fused with basis matmul and
//              radial contraction; emits per-edge scores + v, atomicMax smax.
//              W1/W2 staged into LDS via GLOBAL_LOAD_ASYNC_TO_LDS (ASYNCcnt),
//              overlapped with edge-feature loads and the tmp stage.
//   K2 (edge): exp(score - smax[dst]); atomic denom & unnormalized v segsum.
//   K3 (node): normalize, 16x8 projection, IDX select, bias.
// W1/W2 live in LDS (320KB/WGP) with padded strides for conflict-free b64
// fragment reads. No rw (E x 768) intermediate ever touches HBM.
// ---------------------------------------------------------------------------

typedef float v2f __attribute__((ext_vector_type(2)));
typedef float v8f __attribute__((ext_vector_type(8)));

// Guard only in the DEVICE pass (the host x86 pass never has amdgcn builtins).
#if defined(__HIP_DEVICE_COMPILE__)
#if !__has_builtin(__builtin_amdgcn_wmma_f32_16x16x4_f32)
#error "device: missing __builtin_amdgcn_wmma_f32_16x16x4_f32"
#endif
#if __has_builtin(__builtin_amdgcn_global_load_async_to_lds_b128) && \
    __has_builtin(__builtin_amdgcn_s_wait_asynccnt)
#define EA_HAVE_ASYNC_LDS 1
#else
#define EA_HAVE_ASYNC_LDS 0
#endif
#else
#define EA_HAVE_ASYNC_LDS 0
#endif

#if EA_HAVE_ASYNC_LDS
typedef int ea_v4i __attribute__((vector_size(4 * sizeof(int))));
typedef __attribute__((address_space(3))) ea_v4i ea_lds_v4i;
#endif

__device__ __forceinline__ v8f wmma_f32(v2f a, v2f b, v8f c) {
  return __builtin_amdgcn_wmma_f32_16x16x4_f32(false, a, false, b, (short)0, c,
                                               false, false);
}

__device__ __forceinline__ unsigned int map_f2u(float x) {
  unsigned int u = __float_as_uint(x);
  return (u & 0x80000000u) ? ~u : (u | 0x80000000u);
}
__device__ __forceinline__ float unmap_u2f(unsigned int u) {
  return (u & 0x80000000u) ? __uint_as_float(u & 0x7fffffffu)
                           : __uint_as_float(~u);
}

// --- LDS layout (floats) ---------------------------------------------------
#define W1_STRIDE 36            // 32 cols + pad  (64 rows)
#define W2_STRIDE 68            // 64 cols + pad  (768 rows)
#define W1_FLOATS (64 * W1_STRIDE)       // 2304
#define W2_FLOATS (768 * W2_STRIDE)      // 52224
#define H_STRIDE 68                      // 16 rows
#define T_STRIDE 132                     // tmp: 16 edges x 128 (+4 pad)
#define R_STRIDE 18                      // rw chunk: 16x16 (+2 pad)
#define C_STRIDE 100                     // conv: 16 edges x 96 (+4 pad)
#define PERWAVE_FLOATS (16 * H_STRIDE + 16 * T_STRIDE + 16 * R_STRIDE + 16 * C_STRIDE) // 5088
#define SMEM_FLOATS (W1_FLOATS + W2_FLOATS + 4 * PERWAVE_FLOATS)  // 74880 (~293KB)

__global__ void ea_init_kernel(unsigned int* smax_u, float* denom,
                               float* attn_raw, int N) {
  int i = blockIdx.x * blockDim.x + threadIdx.x;
  if (i < N * 4) { smax_u[i] = 0u; denom[i] = 0.0f; }
  if (i < N * 32) attn_raw[i] = 0.0f;
}

__global__ void __launch_bounds__(128)
ea_edge_kernel(const int* __restrict__ src, const int* __restrict__ dst,
               const float* __restrict__ basis, const float* __restrict__ ef,
               const float* __restrict__ f, const float* __restrict__ W1,
               const float* __restrict__ b1, const float* __restrict__ W2,
               const float* __restrict__ b2, float* __restrict__ ws_scores,
               float* __restrict__ ws_v, unsigned int* __restrict__ smax_u,
               int E) {
  extern __shared__ float smem[];
  float* W1L = smem;
  float* W2L = W1L + W1_FLOATS;

  const int tid = threadIdx.x;
  const int lane = tid & 31;
  const int wave = tid >> 5;
  const int l15 = lane & 15;
  const int hi = lane >> 4;       // which K-half of the fragment this lane holds
  const int kk = 2 * hi;

  float* HL = W2L + W2_FLOATS + wave * PERWAVE_FLOATS;  // 16 x H_STRIDE
  float* TL = HL + 16 * H_STRIDE;                       // 16 x T_STRIDE
  float* RL = TL + 16 * T_STRIDE;                       // 16 x R_STRIDE
  float* CL = RL + 16 * R_STRIDE;                       // 16 x C_STRIDE

  // ---- stage W1 (64x32) and W2 (768x64) into padded LDS -------------------
#if EA_HAVE_ASYNC_LDS
  // Async global->LDS (ASYNCcnt) 16B per lane; no VGPR round trip. Overlapped
  // with the edge-feature loads and the tmp stage below.
  for (int i = tid; i < 512; i += 128) {                 // W1: 512 x 16B
    const float* g = W1 + (size_t)i * 4;
    float* l = W1L + (i >> 3) * W1_STRIDE + (i & 7) * 4;
    __builtin_amdgcn_global_load_async_to_lds_b128(
        (ea_v4i*)g, (ea_lds_v4i*)(void*)l, 0, 0);
  }
  for (int i = tid; i < 12288; i += 128) {               // W2: 12288 x 16B
    const float* g = W2 + (size_t)i * 4;
    float* l = W2L + (i >> 4) * W2_STRIDE + (i & 15) * 4;
    __builtin_amdgcn_global_load_async_to_lds_b128(
        (ea_v4i*)g, (ea_lds_v4i*)(void*)l, 0, 0);
  }
#else
  for (int i = tid; i < 512; i += 128) {                 // W1: 512 float4
    float4 v = ((const float4*)W1)[i];
    float* p = W1L + (i >> 3) * W1_STRIDE + (i & 7) * 4;
    p[0] = v.x; p[1] = v.y; p[2] = v.z; p[3] = v.w;
  }
  for (int i = tid; i < 12288; i += 128) {               // W2: 12288 float4
    float4 v = ((const float4*)W2)[i];
    float* p = W2L + (i >> 4) * W2_STRIDE + (i & 15) * 4;
    p[0] = v.x; p[1] = v.y; p[2] = v.z; p[3] = v.w;
  }
#endif

  const long ebase = (long)(blockIdx.x * 4 + wave) * 16;

  // ---- independent of weights: A-fragments of edge_feats ------------------
  // A tile: edge_feats rows (16 edges x 32). A-frag k0: lane row=l15, K=4k0+kk.
  long eA = ebase + l15; if (eA > E - 1) eA = E - 1;
  const float* X = ef + (size_t)eA * 32;
  v2f A1[8];
#pragma unroll
  for (int k0 = 0; k0 < 8; ++k0) {
    A1[k0].x = X[4 * k0 + kk];
    A1[k0].y = X[4 * k0 + kk + 1];
  }

  // ---- independent of weights: tmp = f[src] @ basis -----------------------
  // lane pair owns one edge: el = lane>>1, h2 selects basis-col half (8 cols).
  const int el = lane >> 1;
  const int h2 = lane & 1;
  long etl = ebase + el;
  const bool valid = (etl < (long)E);
  long et = valid ? etl : (long)(E - 1);
  const int sidx = src[et];
  const float* FR = f + (size_t)sidx * 32;
  const float* BR = basis + (size_t)et * 64 + h2 * 8;
  float bs[4][8];
#pragma unroll
  for (int d = 0; d < 4; ++d)
#pragma unroll
    for (int j = 0; j < 8; ++j) bs[d][j] = BR[d * 16 + j];
#pragma unroll
  for (int m1 = 0; m1 < 8; ++m1) {
    float fm0 = FR[m1 * 4 + 0], fm1 = FR[m1 * 4 + 1];
    float fm2 = FR[m1 * 4 + 2], fm3 = FR[m1 * 4 + 3];
#pragma unroll
    for (int rr = 0; rr < 2; ++rr) {   // k = h2*8 + rr*4 + dd -> m = m1*4+2*h2+rr
      float* tp = TL + el * T_STRIDE + (m1 * 4 + 2 * h2 + rr) * 4;
#pragma unroll
      for (int dd = 0; dd < 4; ++dd) {
        int j = rr * 4 + dd;
        tp[dd] = fm0 * bs[0][j] + fm1 * bs[1][j] + fm2 * bs[2][j] + fm3 * bs[3][j];
      }
    }
  }

  // ---- weights must be resident in LDS beyond this point ------------------
#if EA_HAVE_ASYNC_LDS
  __builtin_amdgcn_s_wait_asynccnt(0);
#endif
  __syncthreads();

  // =========================== MLP1 via WMMA ===============================
#pragma unroll
  for (int n0 = 0; n0 < 4; ++n0) {
    v8f acc = {0.f, 0.f, 0.f, 0.f, 0.f, 0.f, 0.f, 0.f};
#pragma unroll
    for (int k0 = 0; k0 < 8; ++k0) {
      const float* wp = W1L + (n0 * 16 + l15) * W1_STRIDE + 4 * k0 + kk;
      v2f b; b.x = wp[0]; b.y = wp[1];
      acc = wmma_f32(A1[k0], b, acc);
    }
    float bias = b1[n0 * 16 + l15];
#pragma unroll
    for (int r = 0; r < 8; ++r) {               // C layout: row=r+8*hi, col=l15
      float h = acc[r] + bias;
      h = h > 0.f ? h : 0.f;
      HL[(r + 8 * hi) * H_STRIDE + n0 * 16 + l15] = h;
    }
  }

  // ============== MLP2 via WMMA, fused radial contraction ==================
  // rw chunk (16 edges x 16 cols of 768) -> stage in RL -> conv accumulation.
  // cols n0c*16+c all share o = n0c>>1, m = (n0c&1)*16 + c.
  for (int n0c = 0; n0c < 48; ++n0c) {
    v8f acc = {0.f, 0.f, 0.f, 0.f, 0.f, 0.f, 0.f, 0.f};
#pragma unroll
    for (int k0 = 0; k0 < 16; ++k0) {
      const float* ha = HL + l15 * H_STRIDE + 4 * k0 + kk;
      const float* wb = W2L + (n0c * 16 + l15) * W2_STRIDE + 4 * k0 + kk;
      v2f a, b;
      a.x = ha[0]; a.y = ha[1];
      b.x = wb[0]; b.y = wb[1];
      acc = wmma_f32(a, b, acc);
    }
    float bias2 = b2[n0c * 16 + l15];
#pragma unroll
    for (int r = 0; r < 8; ++r)
      RL[(r + 8 * hi) * R_STRIDE + l15] = acc[r] + bias2;

    // consume: lane owns (edge el, d-half h2). DS ops in-order within a wave.
    const int o = n0c >> 1;
    const int mb = (n0c & 1) * 16;
    float c0 = 0.f, c1 = 0.f;
#pragma unroll
    for (int c = 0; c < 16; ++c) {
      float rwv = RL[el * R_STRIDE + c];
      const float* tp = TL + el * T_STRIDE + (mb + c) * 4 + 2 * h2;
      c0 += rwv * tp[0];
      c1 += rwv * tp[1];
    }
    float* cp = CL + el * C_STRIDE + o * 4 + 2 * h2;
    if ((n0c & 1) == 0) { cp[0] = c0; cp[1] = c1; }
    else                { cp[0] += c0; cp[1] += c1; }
  }

  // ===================== attention scores + outputs ========================
  // conv rows: o=0..7 -> k, 8..15 -> q, 16..23 -> v. scores per 4 heads.
  float sc[4];
#pragma unroll
  for (int hh = 0; hh < 4; ++hh) {
    float s = 0.f;
#pragma unroll
    for (int jj = 0; jj < 2; ++jj) {
      int m2 = 2 * hh + jj;
      const float* kp = CL + el * C_STRIDE + m2 * 4;
      const float* qp = CL + el * C_STRIDE + (8 + m2) * 4;
      s += kp[0] * qp[0] + kp[1] * qp[1] + kp[2] * qp[2] + kp[3] * qp[3];
    }
    s *= 0.17677669529663687f;            // HIDDEN^-0.5 = 1/sqrt(32)
    sc[hh] = (s > 0.f) ? s : 0.2f * s;    // leaky_relu(0.2)
  }
  if (valid) {
    int dn = dst[et];
#pragma unroll
    for (int t = 0; t < 2; ++t) {         // this lane handles heads 2*h2+t
      int hh = 2 * h2 + t;
      ws_scores[etl * 4 + hh] = sc[hh];
      atomicMax(&smax_u[dn * 4 + hh], map_f2u(sc[hh]));
    }
#pragma unroll
    for (int o2 = 0; o2 < 8; ++o2) {      // v half owned by this lane
      const float* vp = CL + el * C_STRIDE + (16 + o2) * 4 + 2 * h2;
      float2 vv; vv.x = vp[0]; vv.y = vp[1];
      *(float2*)&ws_v[etl * 32 + o2 * 4 + 2 * h2] = vv;
    }
  }
}

__global__ void ea_softmax_kernel(const int* __restrict__ dst,
                                  const float* __restrict__ ws_scores,
                                  const float* __restrict__ ws_v,
                                  const unsigned int* __restrict__ smax_u,
                                  float* __restrict__ denom,
                                  float* __restrict__ attn_raw, int E) {
  long i = (long)blockIdx.x * blockDim.x + threadIdx.x;
  if (i >= (long)E * 4) return;
  long e = i >> 2;
  int hh = (int)(i & 3);
  int dn = dst[e];
  float smax = unmap_u2f(smax_u[dn * 4 + hh]);
  float ex = expf(ws_scores[e * 4 + hh] - smax);
  atomicAdd(&denom[dn * 4 + hh], ex);
#pragma unroll
  for (int j = 0; j < 8; ++j)
    atomicAdd(&attn_raw[(size_t)dn * 32 + hh * 8 + j],
              ex * ws_v[e * 32 + hh * 8 + j]);
}

__global__ void ea_node_kernel(const float* __restrict__ attn_raw,
                               const float* __restrict__ denom,
                               const float* __restrict__ proj_w,
                               const float* __restrict__ proj_b,
                               float* __restrict__ out, int N) {
  int n = blockIdx.x * blockDim.x + threadIdx.x;
  if (n >= N) return;
  float dnm[4];
#pragma unroll
  for (int hh = 0; hh < 4; ++hh) dnm[hh] = denom[n * 4 + hh];
  float a[8][4];
#pragma unroll
  for (int m = 0; m < 8; ++m)
#pragma unroll
    for (int d = 0; d < 4; ++d) {
      int fl = m * 4 + d;
      float dv = dnm[fl >> 3];
      a[m][d] = (dv != 0.f) ? attn_raw[(size_t)n * 32 + fl] / dv : 0.f;
    }
#pragma unroll
  for (int m = 0; m < 8; ++m) {
#pragma unroll
    for (int d = 0; d < 4; ++d) {
      int o = (d == 0) ? m : (8 + m);     // IDX = [0,1,1,1]
      float s = 0.f;
#pragma unroll
      for (int mp = 0; mp < 8; ++mp) s += proj_w[o * 8 + mp] * a[mp][d];
      if (d == 0) s += proj_b[m];
      out[(size_t)n * 32 + m * 4 + d] = s;
    }
  }
}

extern "C" void kernel_launch(void* const* d_in, const int* in_sizes, int n_in,
                              void* d_out, int out_size, void* d_ws,
                              size_t ws_size, hipStream_t stream) {
  const int* src = (const int*)d_in[0];
  const int* dst = (const int*)d_in[1];
  const float* basis = (const float*)d_in[2];
  const float* ef = (const float*)d_in[3];
  const float* f = (const float*)d_in[4];
  const float* W1 = (const float*)d_in[5];
  const float* b1 = (const float*)d_in[6];
  const float* W2 = (const float*)d_in[7];
  const float* b2 = (const float*)d_in[8];
  const float* proj_w = (const float*)d_in[9];
  const float* proj_b = (const float*)d_in[10];
  const int E = in_sizes[0];
  const int N = in_sizes[4] / 32;

  float* ws = (float*)d_ws;
  float* ws_scores = ws;                              // E*4
  float* ws_v = ws_scores + (size_t)E * 4;            // E*32
  unsigned int* smax_u = (unsigned int*)(ws_v + (size_t)E * 32);  // N*4
  float* denom = (float*)(smax_u + (size_t)N * 4);    // N*4
  float* attn_raw = denom + (size_t)N * 4;            // N*32

  const size_t smem = SMEM_FLOATS * sizeof(float);    // ~293 KB of 320 KB LDS
  (void)hipFuncSetAttribute((const void*)ea_edge_kernel,
                            hipFuncAttributeMaxDynamicSharedMemorySize,
                            (int)smem);

  int initn = N * 32;
  ea_init_kernel<<<(initn + 255) / 256, 256, 0, stream>>>(smax_u, denom,
                                                          attn_raw, N);
  int nblk = (E + 63) / 64;                           // 4 waves x 16 edges
  ea_edge_kernel<<<nblk, 128, smem, stream>>>(src, dst, basis, ef, f, W1, b1,
                                              W2, b2, ws_scores, ws_v, smax_u,
                                              E);
  long t2 = (long)E * 4;
  ea_softmax_kernel<<<(int)((t2 + 255) / 256), 256, 0, stream>>>(
      dst, ws_scores, ws_v, smax_u, denom, attn_raw, E);
  ea_node_kernel<<<(N + 255) / 256, 256, 0, stream>>>(attn_raw, denom, proj_w,
                                                      proj_b, (float*)d_out, N);
}